// GeneratorCombiner_2525440770803
// MI455X (gfx1250) — compile-verified
//
#include <hip/hip_runtime.h>

typedef __attribute__((ext_vector_type(2))) float v2f;
typedef __attribute__((ext_vector_type(8))) float v8f;

#define NQ    10
#define NLAY  5
#define TPB   128

// Map (matmul row r, reduction index kk) -> index in a 1024-amplitude plane.
// gg=0: gate on qubits 0-3 (state bits 9..6)   -> idx = kk*64 + r   (r = low 6 bits)
// gg=1: gate on qubits 4-7 (state bits 5..2)   -> idx = (r/4)*64 + kk*4 + (r%4)
// gg=2: gate on qubits 6-9 (state bits 3..0)   -> idx = r*16 + kk
__device__ __forceinline__ int state_idx(int gg, int r, int kk) {
    if (gg == 0)      return (kk << 6) + r;
    else if (gg == 1) return ((r >> 2) << 6) + (kk << 2) + (r & 3);
    else              return (r << 4) + kk;
}

__global__ __launch_bounds__(TPB) void qgan_wmma_kernel(
    const float* __restrict__ x,    // (BATCH, NQ)
    const float* __restrict__ qp,   // (NGEN, NLAY*NQ*2)
    float* __restrict__ out,        // (BATCH, NGEN*256)
    int ngen)
{
    const int b    = blockIdx.x;    // batch index
    const int g    = blockIdx.y;    // generator index
    const int t    = threadIdx.x;   // 0..127  (4 wave32s)
    const int lane = t & 31;
    const int wave = t >> 5;
    const int half = lane >> 4;     // K half-split for A/B fragments
    const int lm   = lane & 15;     // M for A, N for B/D

    __shared__ float S[2][2048];        // double-buffered state: [buf][plane*1024 + i]
    __shared__ float BR[3][32][16];     // [group][ K | K+16 ][N]: [G_re^T ; -G_im^T]
    __shared__ float BI[3][32][16];     //                          [G_im^T ;  G_re^T]
    __shared__ float ugr[NQ][4];        // per-qubit 2x2 gate (row-major), real
    __shared__ float ugi[NQ][4];        // per-qubit 2x2 gate, imag
    __shared__ float red[TPB];

    // |0...0>
    for (int i = t; i < 2048; i += TPB) S[0][i] = (i == 0) ? 1.0f : 0.0f;
    int cur = 0;

    for (int layer = 0; layer <= NLAY; ++layer) {   // layer 0 = noise embedding (RY only)
        __syncthreads();
        // ---- per-qubit 2x2 gates: U = RZ(phi) * RY(theta) ----
        if (t < NQ) {
            float th, ph;
            if (layer == 0) { th = x[b * NQ + t]; ph = 0.0f; }
            else {
                const float* w = qp + g * (NLAY * NQ * 2) + (layer - 1) * (NQ * 2) + t * 2;
                th = w[0]; ph = w[1];
            }
            float c  = cosf(0.5f * th), s  = sinf(0.5f * th);
            float pc = cosf(0.5f * ph), ps = sinf(0.5f * ph);
            // row0 scaled by e^{-i ph/2} = (pc, -ps); row1 by e^{+i ph/2} = (pc, +ps)
            ugr[t][0] =  pc * c;  ugi[t][0] = -ps * c;   // u00
            ugr[t][1] = -pc * s;  ugi[t][1] =  ps * s;   // u01
            ugr[t][2] =  pc * s;  ugi[t][2] =  ps * s;   // u10
            ugr[t][3] =  pc * c;  ugi[t][3] =  ps * c;   // u11
        }
        __syncthreads();

        // ---- build 16x16 kron gate matrices (B operands, complex-fused along K) ----
        for (int task = t; task < 768; task += TPB) {
            int gg  = task >> 8;
            int rem = task & 255;
            int n = rem >> 4, k = rem & 15;   // G[n][k]: old index k -> new index n
            float re, im;
            if (gg < 2) {
                re = 1.0f; im = 0.0f;
                int q0 = gg << 2;
                #pragma unroll
                for (int j = 0; j < 4; ++j) {
                    int e = (((n >> (3 - j)) & 1) << 1) | ((k >> (3 - j)) & 1);
                    float ur = ugr[q0 + j][e], ui = ugi[q0 + j][e];
                    float nr = re * ur - im * ui;
                    im = re * ui + im * ur;
                    re = nr;
                }
            } else {
                // identity on qubits 6,7; U8 (x) U9 on low 2 bits
                if ((n >> 2) != (k >> 2)) { re = 0.0f; im = 0.0f; }
                else {
                    int e8 = (((n >> 1) & 1) << 1) | ((k >> 1) & 1);
                    int e9 = ((n & 1) << 1) | (k & 1);
                    float ar = ugr[8][e8], ai = ugi[8][e8];
                    float br2 = ugr[9][e9], bi2 = ugi[9][e9];
                    re = ar * br2 - ai * bi2;
                    im = ar * bi2 + ai * br2;
                }
            }
            BR[gg][k][n]      = re;   BR[gg][k + 16][n] = -im;
            BI[gg][k][n]      = im;   BI[gg][k + 16][n] = re;
        }
        __syncthreads();

        // ---- apply the 3 group gates: D(64x16) = A_state(64x32) x B_gate(32x16) ----
        for (int gg = 0; gg < 3; ++gg) {
            const float* src = S[cur];
            float* dst = S[cur ^ 1];
            const int r = (wave << 4) + lm;   // this lane's A-matrix row (tile = wave)
            v8f accr = {0.f,0.f,0.f,0.f,0.f,0.f,0.f,0.f};
            v8f acci = {0.f,0.f,0.f,0.f,0.f,0.f,0.f,0.f};
            #pragma unroll
            for (int kc = 0; kc < 8; ++kc) {
                const int k0 = (kc << 2) + (half << 1);      // even; k0,k0+1 same plane
                const int pl = (k0 >> 4) << 10;              // re plane (0) / im plane (1024)
                v2f a, brf, bif;
                a.x = src[pl + state_idx(gg, r, k0 & 15)];
                a.y = src[pl + state_idx(gg, r, (k0 + 1) & 15)];
                brf.x = BR[gg][k0][lm];     brf.y = BR[gg][k0 + 1][lm];
                bif.x = BI[gg][k0][lm];     bif.y = BI[gg][k0 + 1][lm];
                accr = __builtin_amdgcn_wmma_f32_16x16x4_f32(
                    false, a, false, brf, (short)0, accr, false, false);
                acci = __builtin_amdgcn_wmma_f32_16x16x4_f32(
                    false, a, false, bif, (short)0, acci, false, false);
            }
            #pragma unroll
            for (int v = 0; v < 8; ++v) {
                int rv = (wave << 4) + (half << 3) + v;      // D row (M = v + 8*half)
                int io = state_idx(gg, rv, lm);              // same index map, kk -> n
                dst[io]        = accr[v];
                dst[1024 + io] = acci[v];
            }
            cur ^= 1;
            __syncthreads();
        }

        // ---- CZ entangler chain: diagonal sign flip (skip for noise layer) ----
        if (layer > 0) {
            for (int i = t; i < 1024; i += TPB) {
                if (__popc(i & (i >> 1)) & 1) {
                    S[cur][i]        = -S[cur][i];
                    S[cur][1024 + i] = -S[cur][1024 + i];
                }
            }
            __syncthreads();
        }
    }

    // ---- output: probs[:256] / max(probs[:256])  (the /sum cancels exactly) ----
    __syncthreads();
    {
        float re0 = S[cur][t],       im0 = S[cur][1024 + t];
        float re1 = S[cur][t + 128], im1 = S[cur][1024 + t + 128];
        red[t] = fmaxf(re0 * re0 + im0 * im0, re1 * re1 + im1 * im1);
    }
    __syncthreads();
    for (int s2 = 64; s2 > 0; s2 >>= 1) {
        if (t < s2) red[t] = fmaxf(red[t], red[t + s2]);
        __syncthreads();
    }
    const float inv = 1.0f / red[0];
    const size_t row = (size_t)b * (size_t)(ngen * 256) + (size_t)g * 256;
    for (int p = t; p < 256; p += TPB) {
        float re = S[cur][p], im = S[cur][1024 + p];
        out[row + p] = (re * re + im * im) * inv;
    }
}

extern "C" void kernel_launch(void* const* d_in, const int* in_sizes, int n_in,
                              void* d_out, int out_size, void* d_ws, size_t ws_size,
                              hipStream_t stream) {
    (void)n_in; (void)out_size; (void)d_ws; (void)ws_size;
    const float* x  = (const float*)d_in[0];   // (BATCH, 10) float32
    const float* qp = (const float*)d_in[1];   // (NGEN, 100) float32
    float* out = (float*)d_out;
    int batch = in_sizes[0] / NQ;                // 256
    int ngen  = in_sizes[1] / (NLAY * NQ * 2);   // 4
    dim3 grid(batch, ngen);
    qgan_wmma_kernel<<<grid, TPB, 0, stream>>>(x, qp, out, ngen);
}